// GAT_18537078849649
// MI455X (gfx1250) — compile-verified
//
#include <hip/hip_runtime.h>
#include <hip/hip_bf16.h>

// ---------------------------------------------------------------------------
// 2-layer GAT for MI455X (gfx1250, wave32).
//   GEMMs:        V_WMMA_F32_16X16X4_F32 (full fp32 matrix pipe; GEMMs are
//                 memory-bound at these sizes so fp32 costs nothing extra).
//                 NOUT templated so all W/X loads use immediate offsets.
//   seg-softmax:  3-pass with native u32 atomic-max (order-preserving float
//                 key encoding) and global_atomic_add_f32
//   aggregation:  float4 gather from L2-resident h + 4x atomicAdd per thread
// ---------------------------------------------------------------------------

typedef __attribute__((ext_vector_type(2))) float v2f;
typedef __attribute__((ext_vector_type(8))) float v8f;

#define F_IN   128
#define HEADS  4
#define NEG_SLOPE 0.2f

// ---- order-preserving float <-> u32 key (for native u32 atomic max) --------
__device__ __forceinline__ unsigned f2ord(float f) {
  unsigned u = __float_as_uint(f);
  return (u & 0x80000000u) ? ~u : (u | 0x80000000u);
}
__device__ __forceinline__ float ord2f(unsigned k) {
  unsigned u = (k & 0x80000000u) ? (k & 0x7fffffffu) : ~k;
  return __uint_as_float(u);
}
// key for -inf = ~0xFF800000 = 0x007FFFFF
#define NEG_INF_KEY 0x007FFFFFu

// ---- generic fill ----------------------------------------------------------
__global__ void fill_u32(unsigned* __restrict__ p, unsigned v, long long n) {
  long long i = (long long)blockIdx.x * blockDim.x + threadIdx.x;
  long long stride = (long long)gridDim.x * blockDim.x;
  for (; i < n; i += stride) p[i] = v;
}

// ---- fp32 WMMA GEMM: Hout[M x NOUT] = X[M x 128] @ W[128 x NOUT] -----------
// grid.x = M/16 (M multiple of 16), blockDim.x = 32 * (NOUT/16).
// Each wave computes one 16x16 output tile with 32x V_WMMA_F32_16X16X4_F32.
// NOUT is compile-time so every load in the K loop is base + immediate offset.
template <int NOUT>
__global__ void gemm_wmma_f32(const float* __restrict__ X,
                              const float* __restrict__ W,
                              float* __restrict__ Hout) {
  const int lane = threadIdx.x & 31;
  const int wave = threadIdx.x >> 5;
  const int half = lane >> 4;    // 0: K pair {0,1}; 1: K pair {2,3}
  const int l16  = lane & 15;
  const int row  = blockIdx.x * 16 + l16;       // A-matrix M index (per lane)
  const int col  = wave * 16 + l16;             // B/D-matrix N index (per lane)

  // Per-lane base pointers; all K-loop accesses become immediate offsets.
  const float* xp = X + (size_t)row * F_IN + 2 * half;     // A: K={k,k+1}+2*half
  const float* wp = W + (size_t)(2 * half) * NOUT + col;   // B: rows k+2*half

  v8f acc = {};
#pragma unroll
  for (int k = 0; k < F_IN; k += 4) {
    // A 16x4 fp32 layout: lanes 0-15 hold K={k,k+1}, lanes 16-31 K={k+2,k+3}
    v2f a = *(const v2f*)(xp + k);
    // B 4x16 fp32 layout: lanes 0-15 rows K={k,k+1}, lanes 16-31 K={k+2,k+3}
    v2f b;
    b.x = wp[(size_t)k * NOUT];
    b.y = wp[(size_t)(k + 1) * NOUT];
    acc = __builtin_amdgcn_wmma_f32_16x16x4_f32(
        /*neg_a=*/false, a, /*neg_b=*/false, b,
        /*c_mod=*/(short)0, acc, /*reuse_a=*/false, /*reuse_b=*/false);
  }
  // D layout: VGPR r -> M = r + 8*half, N = l16 (within tile)
  const int mbase = blockIdx.x * 16 + half * 8;
#pragma unroll
  for (int r = 0; r < 8; ++r)
    Hout[(size_t)(mbase + r) * NOUT + col] = acc[r];
}

// ---- per-(node,head) attention coefficients --------------------------------
__global__ void alpha_kernel(const float* __restrict__ Hf,
                             const float* __restrict__ a_src,
                             const float* __restrict__ a_dst,
                             float* __restrict__ as, float* __restrict__ ad,
                             int Nn, int D) {
  int i = blockIdx.x * blockDim.x + threadIdx.x;   // node*HEADS + head
  if (i >= Nn * HEADS) return;
  int n = i / HEADS, h = i % HEADS;
  const float* hp = Hf + (long long)n * HEADS * D + h * D;
  float s = 0.f, d = 0.f;
  for (int k = 0; k < D; ++k) {
    float v = hp[k];
    s += v * a_src[h * D + k];
    d += v * a_dst[h * D + k];
  }
  as[i] = s;
  ad[i] = d;
}

// ---- pass 1: edge logits (leaky relu) + segment max ------------------------
__global__ void edge_logits(const int* __restrict__ srcA,
                            const int* __restrict__ dstA,
                            const float* __restrict__ as,
                            const float* __restrict__ ad,
                            float* __restrict__ eBuf,
                            unsigned* __restrict__ mKey, int E, int Etot) {
  int i = blockIdx.x * blockDim.x + threadIdx.x;   // edge*HEADS + head
  if (i >= Etot * HEADS) return;
  int e = i >> 2, h = i & 3;                       // HEADS == 4
  int s = (e < E) ? srcA[e] : (e - E);             // self loops appended
  int d = (e < E) ? dstA[e] : (e - E);
  float v = as[s * HEADS + h] + ad[d * HEADS + h];
  v = (v > 0.f) ? v : NEG_SLOPE * v;
  eBuf[i] = v;
  atomicMax(&mKey[d * HEADS + h], f2ord(v));       // global_atomic_max_u32
}

// ---- pass 2: exp(e - m) + segment sum --------------------------------------
__global__ void edge_denom(const int* __restrict__ dstA,
                           const float* __restrict__ eBuf,
                           const unsigned* __restrict__ mKey,
                           float* __restrict__ denom, int E, int Etot) {
  int i = blockIdx.x * blockDim.x + threadIdx.x;
  if (i >= Etot * HEADS) return;
  int e = i >> 2, h = i & 3;
  int d = (e < E) ? dstA[e] : (e - E);
  float p = expf(eBuf[i] - ord2f(mKey[d * HEADS + h]));
  atomicAdd(&denom[d * HEADS + h], p);             // global_atomic_add_f32
}

// ---- pass 3: weighted scatter-aggregate ------------------------------------
// work = Etot * (F/4); each thread handles 4 consecutive features of one edge.
__global__ void edge_aggregate(const int* __restrict__ srcA,
                               const int* __restrict__ dstA,
                               const float* __restrict__ Hf,
                               const float* __restrict__ eBuf,
                               const unsigned* __restrict__ mKey,
                               const float* __restrict__ denom,
                               float* __restrict__ out, int E,
                               long long work, int F, int D) {
  long long i = (long long)blockIdx.x * blockDim.x + threadIdx.x;
  if (i >= work) return;
  const int tpe = F >> 2;
  long long e = i / tpe;
  int col = (int)(i % tpe) * 4;
  int h = col / D;
  int s = (e < E) ? srcA[e] : (int)(e - E);
  int d = (e < E) ? dstA[e] : (int)(e - E);
  float m  = ord2f(mKey[d * HEADS + h]);
  float p  = expf(eBuf[e * HEADS + h] - m);
  float al = p / (denom[d * HEADS + h] + 1e-16f);
  float4 hv = *(const float4*)(Hf + (long long)s * F + col);
  float* op = out + (long long)d * F + col;
  atomicAdd(op + 0, hv.x * al);
  atomicAdd(op + 1, hv.y * al);
  atomicAdd(op + 2, hv.z * al);
  atomicAdd(op + 3, hv.w * al);
}

// ---- layer-1 epilogue: bias + ELU (in place) -------------------------------
__global__ void bias_elu(float* __restrict__ buf, const float* __restrict__ b,
                         long long n, int F) {
  long long i = (long long)blockIdx.x * blockDim.x + threadIdx.x;
  if (i >= n) return;
  float v = buf[i] + b[i % F];
  buf[i] = (v > 0.f) ? v : expm1f(v);
}

// ---- layer-2 epilogue: bias + log_softmax over F=64 ------------------------
__global__ void bias_logsoftmax(const float* __restrict__ agg,
                                const float* __restrict__ b,
                                float* __restrict__ out, int Nn, int F) {
  int n = blockIdx.x * blockDim.x + threadIdx.x;
  if (n >= Nn) return;
  const float* ap = agg + (long long)n * F;
  float mx = -INFINITY;
  for (int j = 0; j < F; ++j) mx = fmaxf(mx, ap[j] + b[j]);
  float sum = 0.f;
  for (int j = 0; j < F; ++j) sum += expf(ap[j] + b[j] - mx);
  float ls = logf(sum);
  float* op = out + (long long)n * F;
  for (int j = 0; j < F; ++j) op[j] = ap[j] + b[j] - mx - ls;
}

// ---------------------------------------------------------------------------
extern "C" void kernel_launch(void* const* d_in, const int* in_sizes, int n_in,
                              void* d_out, int out_size, void* d_ws, size_t ws_size,
                              hipStream_t stream) {
  const float* x        = (const float*)d_in[0];
  const int*   ei       = (const int*)d_in[1];
  const float* W1       = (const float*)d_in[2];
  const float* att_src1 = (const float*)d_in[3];
  const float* att_dst1 = (const float*)d_in[4];
  const float* b1       = (const float*)d_in[5];
  const float* W2       = (const float*)d_in[6];
  const float* att_src2 = (const float*)d_in[7];
  const float* att_dst2 = (const float*)d_in[8];
  const float* b2       = (const float*)d_in[9];
  float* outp = (float*)d_out;

  const int N = in_sizes[0] / F_IN;       // 100000 (multiple of 16)
  const int E = in_sizes[1] / 2;          // 1600000
  const int Etot = E + N;                 // self loops appended
  const int HID = 32, NCLS = 16;
  const int F1 = HEADS * HID;             // 128
  const int F2 = HEADS * NCLS;            // 64
  const int* srcA = ei;
  const int* dstA = ei + E;

  // ---- bump-allocate workspace (256B aligned) ----
  char* base = (char*)d_ws;
  size_t off = 0;
  auto alloc = [&](size_t bytes) {
    void* p = base + off;
    off += (bytes + 255) & ~(size_t)255;
    return p;
  };
  float*    h1   = (float*)alloc((size_t)N * F1 * 4);   // layer-1 features
  float*    agg1 = (float*)alloc((size_t)N * F1 * 4);   // layer-1 aggregate / h1post
  float*    h2   = (float*)alloc((size_t)N * F2 * 4);   // layer-2 features
  float*    eBuf = (float*)alloc((size_t)Etot * HEADS * 4); // reused both layers
  float*    as   = (float*)alloc((size_t)N * HEADS * 4);    // reused
  float*    ad   = (float*)alloc((size_t)N * HEADS * 4);    // reused
  unsigned* mKey = (unsigned*)alloc((size_t)N * HEADS * 4); // reused
  float*    den  = (float*)alloc((size_t)N * HEADS * 4);    // reused
  float*    agg2 = h1;  // h1 is dead by the time agg2 is written
  (void)ws_size; (void)n_in; (void)out_size;

  const int TB = 256;
  const int fillB = 2048;
  long long nNH = (long long)N * HEADS;

  // ================= Layer 1 =================
  fill_u32<<<fillB, TB, 0, stream>>>((unsigned*)agg1, 0u, (long long)N * F1);
  fill_u32<<<fillB, TB, 0, stream>>>(mKey, NEG_INF_KEY, nNH);
  fill_u32<<<fillB, TB, 0, stream>>>((unsigned*)den, 0u, nNH);

  gemm_wmma_f32<128><<<N / 16, 32 * (128 / 16), 0, stream>>>(x, W1, h1);
  alpha_kernel<<<(int)((nNH + TB - 1) / TB), TB, 0, stream>>>(
      h1, att_src1, att_dst1, as, ad, N, HID);

  int eh = Etot * HEADS;
  edge_logits<<<(eh + TB - 1) / TB, TB, 0, stream>>>(srcA, dstA, as, ad, eBuf,
                                                     mKey, E, Etot);
  edge_denom<<<(eh + TB - 1) / TB, TB, 0, stream>>>(dstA, eBuf, mKey, den, E,
                                                    Etot);
  long long w1 = (long long)Etot * (F1 / 4);
  edge_aggregate<<<(int)((w1 + TB - 1) / TB), TB, 0, stream>>>(
      srcA, dstA, h1, eBuf, mKey, den, agg1, E, w1, F1, HID);

  long long nF1 = (long long)N * F1;
  bias_elu<<<(int)((nF1 + TB - 1) / TB), TB, 0, stream>>>(agg1, b1, nF1, F1);

  // ================= Layer 2 =================
  fill_u32<<<fillB, TB, 0, stream>>>((unsigned*)agg2, 0u, (long long)N * F2);
  fill_u32<<<fillB, TB, 0, stream>>>(mKey, NEG_INF_KEY, nNH);
  fill_u32<<<fillB, TB, 0, stream>>>((unsigned*)den, 0u, nNH);

  gemm_wmma_f32<64><<<N / 16, 32 * (64 / 16), 0, stream>>>(agg1, W2, h2);
  alpha_kernel<<<(int)((nNH + TB - 1) / TB), TB, 0, stream>>>(
      h2, att_src2, att_dst2, as, ad, N, NCLS);

  edge_logits<<<(eh + TB - 1) / TB, TB, 0, stream>>>(srcA, dstA, as, ad, eBuf,
                                                     mKey, E, Etot);
  edge_denom<<<(eh + TB - 1) / TB, TB, 0, stream>>>(dstA, eBuf, mKey, den, E,
                                                    Etot);
  long long w2 = (long long)Etot * (F2 / 4);
  edge_aggregate<<<(int)((w2 + TB - 1) / TB), TB, 0, stream>>>(
      srcA, dstA, h2, eBuf, mKey, den, agg2, E, w2, F2, NCLS);

  bias_logsoftmax<<<(N + TB - 1) / TB, TB, 0, stream>>>(agg2, b2, outp, N, F2);
}